// RRoPE_87995289960714
// MI455X (gfx1250) — compile-verified
//
#include <hip/hip_runtime.h>

typedef __attribute__((ext_vector_type(2))) float v2f;
typedef __attribute__((ext_vector_type(4))) float v4f;
typedef __attribute__((ext_vector_type(8))) float v8f;

#define SEQ  2048
#define DK   128
#define NROW 128   // B*H = 8*16 rows share one seq position
#define LROW 132   // padded floats per LDS row (bank-conflict-free A reads)

#define AS1 __attribute__((address_space(1)))
#define AS3 __attribute__((address_space(3)))

// Parameter type of the async-LDS builtin per hipcc's diagnostic:
// 'int __attribute__((__vector_size__(4 * sizeof(int)))) *'
typedef int av4i __attribute__((__vector_size__(4 * sizeof(int))));
typedef AS1 av4i* gptr_v4i;   // pointee in addrspace(1)
typedef AS3 av4i* lptr_v4i;   // pointee in addrspace(3)

#if __has_builtin(__builtin_amdgcn_global_load_async_to_lds_b128)
#define HAVE_ASYNC_LDS 1
#else
#define HAVE_ASYNC_LDS 0
#endif

// out[row, s, o] = sum_i x[row, s, i] * R[pos[s], o, i], R block-diagonal 2x2.
// One block per s; wave w handles rows 16w..16w+15; 8 diagonal 16x16 tiles,
// each as 4 chained V_WMMA_F32_16X16X4_F32. x is staged to LDS via the
// gfx1250 async global->LDS DMA path when the toolchain exposes it.
__global__ __launch_bounds__(256) void rope_wmma_kernel(
    const float* __restrict__ x,
    const int*   __restrict__ posv,
    const float* __restrict__ rmat,
    float*       __restrict__ out)
{
    extern __shared__ float xs[];          // NROW * LROW floats

    const int tid  = threadIdx.x;
    const int s    = blockIdx.x;
    const int lane = tid & 31;
    const int wave = tid >> 5;
    const int l15  = lane & 15;            // A row-in-tile == B col-in-tile
    const int hi   = lane >> 4;            // half-wave selects K sub-pair

    // ---- Stage x[:, :, s, :] into LDS: 128 rows x 512B ----
#if HAVE_ASYNC_LDS
    for (int i = 0; i < 16; ++i) {
        int idx = tid + (i << 8);          // 0..4095 float4 chunks
        int row = idx >> 5;                // 32 float4 per row -> 512B/instr
        int c4  = idx & 31;
        __builtin_amdgcn_global_load_async_to_lds_b128(
            (gptr_v4i)(x + ((size_t)row * SEQ + s) * DK + (c4 << 2)),
            (lptr_v4i)(xs + row * LROW + (c4 << 2)),
            0, 0);
    }
#if __has_builtin(__builtin_amdgcn_s_wait_asynccnt)
    __builtin_amdgcn_s_wait_asynccnt(0);
#else
    asm volatile("s_wait_asynccnt 0x0" ::: "memory");
#endif
#else
    for (int i = 0; i < 16; ++i) {
        int idx = tid + (i << 8);
        int row = idx >> 5;
        int c4  = idx & 31;
        v4f val = *(const v4f*)(x + ((size_t)row * SEQ + s) * DK + (c4 << 2));
        *(v4f*)(xs + row * LROW + (c4 << 2)) = val;
    }
#endif
    __syncthreads();

    const int    p  = posv[s];
    const float* rb = rmat + (size_t)p * DK * DK;

    for (int tt = 0; tt < 8; ++tt) {       // 16x16 diagonal tile of R
        v8f acc = {};
#pragma unroll
        for (int cc = 0; cc < 4; ++cc) {   // K = 16 in 4 chunks of 4
            const int k0 = tt * 16 + cc * 4 + hi * 2;   // absolute i index

            // A fragment: lane holds row (wave*16 + l15), K pair (k0, k0+1)
            v2f a = *(const v2f*)(xs + (wave * 16 + l15) * LROW + k0);

            // B fragment: B[k, n] = R[tt*16 + n, k]; lane col n = l15.
            v2f b = *(const v2f*)(rb + (size_t)(tt * 16 + l15) * DK + k0);
            // Zero slots off the 2x2 pair diagonal (pair(k) != pair(n)).
            if ((cc * 2 + hi) != (l15 >> 1)) { b.x = 0.0f; b.y = 0.0f; }

            // D = A(16x4) * B(4x16) + C, f32 accumulate
            acc = __builtin_amdgcn_wmma_f32_16x16x4_f32(
                false, a, false, b, (short)0, acc, false, false);
        }

        // D layout: VGPR vv -> row (vv + 8*hi), col l15; 64B contiguous/half
#pragma unroll
        for (int vv = 0; vv < 8; ++vv) {
            int row = wave * 16 + (hi << 3) + vv;
            out[((size_t)row * SEQ + s) * DK + tt * 16 + l15] = acc[vv];
        }
    }
}

extern "C" void kernel_launch(void* const* d_in, const int* in_sizes, int n_in,
                              void* d_out, int out_size, void* d_ws, size_t ws_size,
                              hipStream_t stream) {
    (void)in_sizes; (void)n_in; (void)d_ws; (void)ws_size; (void)out_size;
    const float* x    = (const float*)d_in[0];
    const int*   tp   = (const int*)d_in[1];   // token_positions (int32 under JAX defaults)
    const float* rmat = (const float*)d_in[2];
    float*       out  = (float*)d_out;

    const size_t lds_bytes = (size_t)NROW * LROW * sizeof(float);  // 67,584 B
    rope_wmma_kernel<<<dim3(SEQ), dim3(256), lds_bytes, stream>>>(x, tp, rmat, out);
}